// GDL_67929202753539
// MI455X (gfx1250) — compile-verified
//
#include <hip/hip_runtime.h>

typedef __attribute__((ext_vector_type(2))) float v2f;
typedef __attribute__((ext_vector_type(4))) float v4f;
typedef __attribute__((ext_vector_type(8))) float v8f;

#define GDL_H    512
#define GDL_W    512
#define GDL_IMGS 96
#define GDL_N    (GDL_IMGS * GDL_H * GDL_W)   // 25,165,824 elements per tensor
#define BLOCKS1  6144                          // 6144*256 threads, 1 vec4-quad each

// Exact f32 reduction of all 32 lanes of a wave using V_WMMA_F32_16X16X4_F32.
// A-layout (16x4 f32): lanes 0-15 -> A[lane][0] (VGPR0), lanes 16-31 -> A[lane-16][2].
// With B = ones, D[m][n] = s_m + s_{m+16}. D-layout: lanes 0-15 hold rows 0..7 in
// VGPR0..7, lanes 16-31 hold rows 8..15. Sum the 8 VGPRs then fold lane^16.
// Multiplies are by 1.0 (exact); accumulation is f32 RNE -> deterministic.
__device__ __forceinline__ float wave_reduce32(float s) {
  v2f a; a[0] = s;    a[1] = 0.0f;
  v2f b; b[0] = 1.0f; b[1] = 1.0f;
  v8f c = {0.f, 0.f, 0.f, 0.f, 0.f, 0.f, 0.f, 0.f};
  v8f d = __builtin_amdgcn_wmma_f32_16x16x4_f32(false, a, false, b,
                                                (short)0, c, false, false);
  float t = d[0] + d[1] + d[2] + d[3] + d[4] + d[5] + d[6] + d[7];
  t += __shfl_xor(t, 16, 32);   // combine rows 0..7(+16..23) with 8..15(+24..31)
  return t;                      // full 32-lane sum, broadcast to all lanes
}

__device__ __forceinline__ float block_reduce(float local, float* smem) {
  float wsum = wave_reduce32(local);
  const int lane = threadIdx.x & 31;
  const int wave = threadIdx.x >> 5;
  if (lane == 0) smem[wave] = wsum;
  __syncthreads();
  float bsum = 0.0f;
  if (threadIdx.x < 32) {                       // wave-uniform: full-EXEC wave 0
    float v = (lane < 8) ? smem[lane] : 0.0f;
    bsum = wave_reduce32(v);
  }
  return bsum;                                  // valid in wave 0
}

__global__ void __launch_bounds__(256) gdl_partial_kernel(
    const float* __restrict__ yt, const float* __restrict__ yp,
    float* __restrict__ partial) {
  const int tid = blockIdx.x * 256 + threadIdx.x;
  // tid -> (img, h in [0,256), wv in [0,64)); each thread owns one float4 quad
  const int img = tid >> 14;          // / (256*64)
  const int r   = tid & 16383;
  const int h   = r >> 6;
  const int wv  = r & 63;

  const int base = img * (GDL_H * GDL_W);
  const int row0 = base + h * GDL_W;
  const int row1 = base + (GDL_H - 1 - h) * GDL_W;
  const int c0   = 4 * wv;                    // w  = c0 + j
  const int c1   = (GDL_W - 4) - 4 * wv;      // wm = c1 + (3 - j)

  // 8 x 128-bit coalesced, non-temporal (single-use streaming) loads
  const v4f t00 = __builtin_nontemporal_load((const v4f*)(yt + row0 + c0));
  const v4f t01 = __builtin_nontemporal_load((const v4f*)(yt + row0 + c1));
  const v4f t10 = __builtin_nontemporal_load((const v4f*)(yt + row1 + c0));
  const v4f t11 = __builtin_nontemporal_load((const v4f*)(yt + row1 + c1));
  const v4f p00 = __builtin_nontemporal_load((const v4f*)(yp + row0 + c0));
  const v4f p01 = __builtin_nontemporal_load((const v4f*)(yp + row0 + c1));
  const v4f p10 = __builtin_nontemporal_load((const v4f*)(yp + row1 + c0));
  const v4f p11 = __builtin_nontemporal_load((const v4f*)(yp + row1 + c1));

  float local = 0.0f;
#pragma unroll
  for (int j = 0; j < 4; ++j) {
    const int jm = 3 - j;   // mirrored element inside the reversed float4
    // t1 at (h,w) [== t1 at (hm,w)]
    float d1a = fabsf(t00[j]  - t10[j])  - fabsf(p00[j]  - p10[j]);
    // t1 at (h,wm) [== t1 at (hm,wm)]
    float d1b = fabsf(t01[jm] - t11[jm]) - fabsf(p01[jm] - p11[jm]);
    // t2 at (h,w) [== t2 at (h,wm)]
    float d2a = fabsf(t01[jm] - t00[j])  - fabsf(p01[jm] - p00[j]);
    // t2 at (hm,w) [== t2 at (hm,wm)]
    float d2b = fabsf(t11[jm] - t10[j])  - fabsf(p11[jm] - p10[j]);
    local += d1a * d1a + d1b * d1b + d2a * d2a + d2b * d2b;
  }
  // (overall factor of 2 for the symmetry is applied in the final kernel)

  __shared__ float smem[8];
  float bsum = block_reduce(local, smem);
  if (threadIdx.x == 0) partial[blockIdx.x] = bsum;
}

__global__ void __launch_bounds__(256) gdl_final_kernel(
    const float* __restrict__ partial, float* __restrict__ out, int n) {
  float local = 0.0f;
  for (int i = threadIdx.x; i < n; i += 256) local += partial[i];
  __shared__ float smem[8];
  float bsum = block_reduce(local, smem);
  if (threadIdx.x == 0)
    out[0] = bsum * (2.0f / (float)GDL_N);   // x2 symmetry factor, / N for mean
}

extern "C" void kernel_launch(void* const* d_in, const int* in_sizes, int n_in,
                              void* d_out, int out_size, void* d_ws, size_t ws_size,
                              hipStream_t stream) {
  const float* y_true = (const float*)d_in[0];
  const float* y_pred = (const float*)d_in[1];
  float* partial = (float*)d_ws;               // BLOCKS1 floats = 24 KB scratch
  (void)in_sizes; (void)n_in; (void)out_size; (void)ws_size;

  gdl_partial_kernel<<<BLOCKS1, 256, 0, stream>>>(y_true, y_pred, partial);
  gdl_final_kernel<<<1, 256, 0, stream>>>(partial, (float*)d_out, BLOCKS1);
}